// NaiveNaiveBayes_1451698946185
// MI455X (gfx1250) — compile-verified
//
#include <hip/hip_runtime.h>
#include <cstdint>
#include <cstddef>

// NaiveNaiveBayes: out[n] = prod_d fixup(exp(A[d]*x^2 + B[d]*x + C[d]))
// Memory-bound streamer: 512MB in / 2MB out -> ~22us floor @ 23.3 TB/s.
// CDNA5 path: global_load_async_to_lds_b128 (ASYNCcnt) 4-deep ring buffer per
// wave, ds_load_b128 reads, v_exp_f32 via exp2 coefficients, wave32 shuffle
// product reduction. All wave-uniform control flow forced scalar via
// readfirstlane so the hot loop is SALU-controlled (co-issues with VALU).

#define D_COLS 256
#define WAVES_PER_BLOCK 8
#define NBUF 4           // ring buffer depth (3 rows prefetched ahead)
#define GRID_BLOCKS 2048 // 16384 waves -> 32 rows/wave at N=524288

__device__ __forceinline__ float nb_term(float x, float A, float B, float C) {
  // t = log2( exp(logN(x;m,s) - logN(x;0,1)) )
  float t = __builtin_fmaf(__builtin_fmaf(A, x, B), x, C);
#if __has_builtin(__builtin_amdgcn_exp2f)
  float r = __builtin_amdgcn_exp2f(t); // v_exp_f32
#else
  float r = __builtin_exp2f(t);
#endif
  float o = fmaxf(r, 0.01f);
  // reference: max(exp,0.01) then (nan|posinf)->100.  r<=FLT_MAX is false for
  // both NaN and +inf, so this reproduces both fixups.
  return (r <= 3.402823466e+38f) ? o : 100.0f;
}

// Issue one 1KB row (256 fp32) into this wave's LDS slot via the CDNA5 async
// data mover path. INST offset applies to both global and LDS addresses, so a
// single offset:512 covers the second half of the row. s_wait_dscnt 0 makes
// buffer reuse safe (prior ds reads of this slot are complete before reissue).
__device__ __forceinline__ void issue_async_row(uint32_t ldsoff, const float* gbase, int lane) {
  uint64_t ga = (uint64_t)(uintptr_t)gbase + (uint32_t)(lane * 16);
  asm volatile(
      "s_wait_dscnt 0\n\t"
      "global_load_async_to_lds_b128 %0, %1, off\n\t"
      "global_load_async_to_lds_b128 %0, %1, off offset:512"
      :
      : "v"(ldsoff), "v"(ga)
      : "memory");
}

__device__ __forceinline__ float row_product(const float* lbuf, int lane,
                                             const float* A, const float* B, const float* C) {
  const float4 v0 = ((const float4*)lbuf)[lane];      // cols 4l..4l+3   (ds_load_b128)
  const float4 v1 = ((const float4*)lbuf)[lane + 32]; // cols 128+4l..+3 (ds_load_b128)
  const float xr[8] = {v0.x, v0.y, v0.z, v0.w, v1.x, v1.y, v1.z, v1.w};
  float p0 = 1.0f, p1 = 1.0f; // two chains for ILP against exp2 latency
#pragma unroll
  for (int j = 0; j < 4; ++j) p0 *= nb_term(xr[j], A[j], B[j], C[j]);
#pragma unroll
  for (int j = 4; j < 8; ++j) p1 *= nb_term(xr[j], A[j], B[j], C[j]);
  float p = p0 * p1;
#pragma unroll
  for (int off = 16; off > 0; off >>= 1) // wave32 product reduction
    p *= __shfl_xor(p, off, 32);
  return p;
}

__global__ __launch_bounds__(256) void nbayes_kernel(const float* __restrict__ x,
                                                     const float* __restrict__ means,
                                                     const float* __restrict__ stds,
                                                     float* __restrict__ out, int nrows) {
  __shared__ float buf[WAVES_PER_BLOCK][NBUF][D_COLS]; // 32 KB/block
  const int lane = (int)(threadIdx.x & 31u);
  // wave id is uniform within the wave: force it into an SGPR so all row/loop
  // control below is scalar (s_cmp/s_cbranch), not EXEC-masked VALU.
  const int wid = __builtin_amdgcn_readfirstlane((int)(threadIdx.x >> 5));

  // --- one-time per-lane coefficients (log2 domain) for this lane's 8 cols ---
  const float4 m0 = ((const float4*)means)[lane];
  const float4 s0 = ((const float4*)stds)[lane];
  const float4 m1 = ((const float4*)means)[lane + 32];
  const float4 s1 = ((const float4*)stds)[lane + 32];
  const float mm[8] = {m0.x, m0.y, m0.z, m0.w, m1.x, m1.y, m1.z, m1.w};
  const float ss[8] = {s0.x, s0.y, s0.z, s0.w, s1.x, s1.y, s1.z, s1.w};
  float A[8], B[8], C[8];
  const float L = 1.4426950408889634f; // log2(e)
#pragma unroll
  for (int j = 0; j < 8; ++j) {
    float inv = 1.0f / ss[j];
    float inv2 = inv * inv;
    A[j] = 0.5f * L * (1.0f - inv2);
    B[j] = L * mm[j] * inv2;
    C[j] = -L * (__logf(ss[j]) + 0.5f * mm[j] * mm[j] * inv2);
  }

  const int gw = (int)(blockIdx.x * WAVES_PER_BLOCK) + wid; // scalar global wave id
  const int nw = (int)(gridDim.x * WAVES_PER_BLOCK);        // scalar total waves

  // 0-based LDS byte offset of this wave's slot (low 32 bits of flat shared addr)
  const uint32_t lbase = (uint32_t)(uintptr_t)(&buf[wid][0][0]) + (uint32_t)(lane * 16);

  // --- prologue: prefetch up to NBUF-1 rows (scalar guards) ---
#pragma unroll
  for (int k = 0; k < NBUF - 1; ++k) {
    int r = gw + k * nw;
    if (r < nrows)
      issue_async_row(lbase + (uint32_t)(k * D_COLS * 4), x + (size_t)r * D_COLS, lane);
  }

  int row = gw;
  int i = 0;
  // --- main loop: keep 3 rows (6 async b128s) in flight per wave ---
  while (row + (NBUF - 1) * nw < nrows) {
    issue_async_row(lbase + (uint32_t)(((i + NBUF - 1) & (NBUF - 1)) * D_COLS * 4),
                    x + (size_t)(row + (NBUF - 1) * nw) * D_COLS, lane);
    asm volatile("s_wait_asynccnt 6" ::: "memory"); // oldest row landed (in-order done)
    float p = row_product(&buf[wid][i & (NBUF - 1)][0], lane, A, B, C);
    if (lane == 0) out[row] = p;
    row += nw;
    ++i;
  }
  // --- epilogue: drain the <=3 buffered rows ---
  asm volatile("s_wait_asynccnt 0" ::: "memory");
  while (row < nrows) {
    float p = row_product(&buf[wid][i & (NBUF - 1)][0], lane, A, B, C);
    if (lane == 0) out[row] = p;
    row += nw;
    ++i;
  }
}

extern "C" void kernel_launch(void* const* d_in, const int* in_sizes, int n_in,
                              void* d_out, int out_size, void* d_ws, size_t ws_size,
                              hipStream_t stream) {
  (void)n_in;
  (void)d_ws;
  (void)ws_size;
  (void)out_size;
  const float* x = (const float*)d_in[0];
  const float* means = (const float*)d_in[1];
  const float* stds = (const float*)d_in[2];
  float* out = (float*)d_out;
  const int nrows = in_sizes[0] / D_COLS; // 524288
  dim3 grid(GRID_BLOCKS), block(WAVES_PER_BLOCK * 32);
  hipLaunchKernelGGL(nbayes_kernel, grid, block, 0, stream, x, means, stds, out, nrows);
}